// RelationModule_43404939493427
// MI455X (gfx1250) — compile-verified
//
#include <hip/hip_runtime.h>
#include <hip/hip_bf16.h>
#include <math.h>

// ---------------------------------------------------------------------------
// RelationModule on MI455X (gfx1250, wave32, WMMA).
// Memory-bound workload: dominated by the 402 MB position_embedding stream.
// All matmuls use native f32 WMMA (V_WMMA_F32_16X16X4_F32) - no precision
// loss vs the f32 reference, and HBM (23.3 TB/s) is the limiter anyway.
// ---------------------------------------------------------------------------

#define B_   16
#define K_   256
#define APP_ 768
#define DK_  96
#define NR_  8
#define M_   (B_ * K_)     // 4096 rows of f_a
#define J_   (NR_ * DK_)   // 768 fused head-feature columns

typedef __attribute__((ext_vector_type(2))) float v2f;
typedef __attribute__((ext_vector_type(8))) float v8f;

__device__ __forceinline__ v8f wmma_f32_16x16x4(v2f a, v2f b, v8f c) {
    // D = A(16x4) * B(4x16) + C(16x16), all f32, wave32.
    return __builtin_amdgcn_wmma_f32_16x16x4_f32(
        /*neg_a=*/false, a, /*neg_b=*/false, b,
        /*c_mod=*/(short)0, c, /*reuse_a=*/false, /*reuse_b=*/false);
}

// ---------------------------------------------------------------------------
// Kernel 1: projections.  out[m, j] = sum_d f_a[m,d] * W[j,d] (+ bias[j]),
// with j = r*DK + k.  w_q is additionally pre-scaled by 1/sqrt(DK).
// One wave per 16x16 output tile; 192 steps of WMMA 16x16x4 over APP=768.
// ---------------------------------------------------------------------------
__global__ __launch_bounds__(128) void proj_kernel(
    const float* __restrict__ f_a,
    const float* __restrict__ Wk, const float* __restrict__ bk,
    const float* __restrict__ Wq, const float* __restrict__ bq,
    const float* __restrict__ Wv, const float* __restrict__ bv,
    float* __restrict__ outk, float* __restrict__ outq, float* __restrict__ outv)
{
    const int lane = threadIdx.x & 31;
    const int wave = threadIdx.x >> 5;
    const int mt   = blockIdx.x;                 // 0..255  (m tile)
    const int jt   = blockIdx.y * 4 + wave;      // 0..47   (j tile)
    const int which = blockIdx.z;                // 0=K 1=Q 2=V

    const float* W;  const float* bias;  float* out;  float scale = 1.0f;
    if (which == 0)      { W = Wk; bias = bk; out = outk; }
    else if (which == 1) { W = Wq; bias = bq; out = outq; scale = 0.1020620726f; } // 1/sqrt(96)
    else                 { W = Wv; bias = bv; out = outv; }

    const int l    = lane & 15;
    const int half = lane >> 4;                  // 0 or 1
    const float* aptr = f_a + (mt * 16 + l) * APP_ + half * 2;  // A row = m
    const float* bptr = W   + (jt * 16 + l) * APP_ + half * 2;  // B col = j (B=W^T)

    v8f acc = {};
    for (int kk = 0; kk < APP_; kk += 4) {
        v2f a = { aptr[kk], aptr[kk + 1] };
        v2f b = { bptr[kk], bptr[kk + 1] };
        acc = wmma_f32_16x16x4(a, b, acc);
    }

    const int j  = jt * 16 + l;                  // D column (N) = lane&15
    const float bj = bias[j];
#pragma unroll
    for (int i = 0; i < 8; ++i) {
        const int m = mt * 16 + i + half * 8;    // D row: vgpr i -> M=i / i+8
        out[m * J_ + j] = (acc[i] + bj) * scale;
    }
}

// ---------------------------------------------------------------------------
// Kernel 2: logits[r,b,m,n] = log(max(pe[b,m,n,:]·WG_w[r,:] + WG_b[r], 1e-6))
//                             + w_k[b,m, r*96:]·w_q[b,n, r*96:]   (q pre-scaled)
// Block = 256 threads, one (b, 16m, 16n) tile, all 8 heads.
// Phase A (VALU): stream the 96-float pe vector per (m,n) -> 8 biases -> LDS.
// Phase B (WMMA): wave r computes the 16x16x96 score GEMM, adds LDS bias.
// ---------------------------------------------------------------------------
__global__ __launch_bounds__(256) void logits_kernel(
    const float* __restrict__ pe,
    const float* __restrict__ WG_w, const float* __restrict__ WG_b,
    const float* __restrict__ wk,   const float* __restrict__ wq,
    float* __restrict__ att)
{
    __shared__ float s_wgw[NR_ * 96];    // 8x96 weights
    __shared__ float s_wgb[NR_];
    __shared__ float s_gb[NR_ * 256];    // log-bias tile [r][m][n]

    const int t  = threadIdx.x;
    const int nt = blockIdx.x;           // 0..15
    const int mt = blockIdx.y;           // 0..15
    const int b  = blockIdx.z;           // 0..15

    for (int i = t; i < NR_ * 96; i += 256) s_wgw[i] = WG_w[i];
    if (t < NR_) s_wgb[t] = WG_b[t];
    __syncthreads();

    // ---- Phase A: geometric bias (this is the 402 MB stream) ----
    {
        const int m = t >> 4, n = t & 15;
        const float* pptr =
            pe + (((size_t)(b * K_ + mt * 16 + m) * K_) + (nt * 16 + n)) * 96;
        float accr[NR_];
#pragma unroll
        for (int r = 0; r < NR_; ++r) accr[r] = s_wgb[r];
        for (int g = 0; g < 96; g += 4) {
            const float4 x = *(const float4*)(pptr + g);
#pragma unroll
            for (int r = 0; r < NR_; ++r) {
                accr[r] = fmaf(x.x, s_wgw[r * 96 + g    ], accr[r]);
                accr[r] = fmaf(x.y, s_wgw[r * 96 + g + 1], accr[r]);
                accr[r] = fmaf(x.z, s_wgw[r * 96 + g + 2], accr[r]);
                accr[r] = fmaf(x.w, s_wgw[r * 96 + g + 3], accr[r]);
            }
        }
#pragma unroll
        for (int r = 0; r < NR_; ++r)    // relu + clip(1e-6) + log == log(max(x,1e-6))
            s_gb[r * 256 + m * 16 + n] = __logf(fmaxf(accr[r], 1e-6f));
    }
    __syncthreads();

    // ---- Phase B: scores via WMMA, wave r handles head r ----
    {
        const int lane = t & 31;
        const int r    = t >> 5;             // 8 waves = 8 heads
        const int l    = lane & 15;
        const int half = lane >> 4;
        const float* kptr = wk + (size_t)(b * K_ + mt * 16 + l) * J_ + r * DK_ + half * 2;
        const float* qptr = wq + (size_t)(b * K_ + nt * 16 + l) * J_ + r * DK_ + half * 2;
        v8f acc = {};
        for (int kk = 0; kk < DK_; kk += 4) {
            v2f a = { kptr[kk], kptr[kk + 1] };   // A[m][k]
            v2f q = { qptr[kk], qptr[kk + 1] };   // B[k][n] = w_q[n][k]
            acc = wmma_f32_16x16x4(a, q, acc);
        }
        float* ab = att + ((size_t)(r * B_ + b) * K_ + mt * 16) * K_ + nt * 16;
#pragma unroll
        for (int i = 0; i < 8; ++i) {
            const int ml = i + half * 8;          // local m
            ab[(size_t)ml * K_ + l] = acc[i] + s_gb[r * 256 + ml * 16 + l];
        }
    }
}

// ---------------------------------------------------------------------------
// Kernel 3: softmax over m (axis 2) of att[r,b,m,n], in place.
// One block per (r,b); thread n walks its column (coalesced across threads).
// ---------------------------------------------------------------------------
__global__ __launch_bounds__(256) void softmax_kernel(float* __restrict__ att)
{
    const int rb = blockIdx.x;                   // 0..127 = r*16+b
    const int n  = threadIdx.x;                  // 0..255
    float* col = att + (size_t)rb * K_ * K_ + n;

    float mx = -3.4e38f, sum = 0.0f;
    for (int m = 0; m < K_; ++m) {
        const float x  = col[(size_t)m * K_];
        const float nx = fmaxf(mx, x);
        sum = sum * __expf(mx - nx) + __expf(x - nx);
        mx  = nx;
    }
    const float inv = 1.0f / sum;
    for (int m = 0; m < K_; ++m) {
        col[(size_t)m * K_] = __expf(col[(size_t)m * K_] - mx) * inv;
    }
}

// ---------------------------------------------------------------------------
// Kernel 4: out[b, n, r*96+k] = sum_m att[r,b,m,n] * w_v[b,m, r*96+k] + f_a.
// One wave per (r, b, 16n, 16k) tile: A = att^T tile (16n x 4m), B = v tile.
// ---------------------------------------------------------------------------
__global__ __launch_bounds__(32) void out_kernel(
    const float* __restrict__ att, const float* __restrict__ wv,
    const float* __restrict__ f_a, float* __restrict__ out)
{
    const int lane = threadIdx.x & 31;
    const int nt = blockIdx.x / 6;               // 0..15
    const int kt = blockIdx.x % 6;               // 0..5
    const int b  = blockIdx.y;                   // 0..15
    const int r  = blockIdx.z;                   // 0..7

    const int l    = lane & 15;
    const int half = lane >> 4;
    const float* ab = att + (size_t)(r * B_ + b) * K_ * K_;
    const float* vb = wv  + (size_t)(b * K_) * J_ + r * DK_;

    v8f acc = {};
    for (int m = 0; m < K_; m += 4) {
        const int kb = m + half * 2;
        v2f a = { ab[(size_t)kb * K_ + nt * 16 + l],          // A[n][c] = att[m+c][n]
                  ab[(size_t)(kb + 1) * K_ + nt * 16 + l] };
        v2f v = { vb[(size_t)kb * J_ + kt * 16 + l],          // B[c][k] = v[m+c][k]
                  vb[(size_t)(kb + 1) * J_ + kt * 16 + l] };
        acc = wmma_f32_16x16x4(a, v, acc);
    }

    const int kcol = r * DK_ + kt * 16 + l;
#pragma unroll
    for (int i = 0; i < 8; ++i) {
        const int n = nt * 16 + i + half * 8;    // D row = n
        const size_t gidx = (size_t)(b * K_ + n) * J_ + kcol;
        out[gidx] = acc[i] + f_a[gidx];          // fused residual
    }
}

// ---------------------------------------------------------------------------
extern "C" void kernel_launch(void* const* d_in, const int* in_sizes, int n_in,
                              void* d_out, int out_size, void* d_ws, size_t ws_size,
                              hipStream_t stream) {
    const float* f_a  = (const float*)d_in[0];
    const float* pe   = (const float*)d_in[1];
    const float* WG_w = (const float*)d_in[2];
    const float* WG_b = (const float*)d_in[3];
    const float* WK_w = (const float*)d_in[4];
    const float* WK_b = (const float*)d_in[5];
    const float* WQ_w = (const float*)d_in[6];
    const float* WQ_b = (const float*)d_in[7];
    const float* WV_w = (const float*)d_in[8];
    const float* WV_b = (const float*)d_in[9];
    float* out = (float*)d_out;

    // Workspace: w_k | w_q | w_v  (each 4096x768 f32)  |  att (8x16x256x256 f32)
    float* wk  = (float*)d_ws;
    float* wq  = wk + (size_t)M_ * J_;
    float* wv  = wq + (size_t)M_ * J_;
    float* att = wv + (size_t)M_ * J_;           // total ~71.3 MB

    // 1) projections: grid (m-tiles, j-tile groups of 4, which), 4 waves/block
    proj_kernel<<<dim3(M_ / 16, (J_ / 16) / 4, 3), 128, 0, stream>>>(
        f_a, WK_w, WK_b, WQ_w, WQ_b, WV_w, WV_b, wk, wq, wv);

    // 2) logits = log-bias + scores   (streams the 402 MB pe tensor)
    logits_kernel<<<dim3(K_ / 16, K_ / 16, B_), 256, 0, stream>>>(
        pe, WG_w, WG_b, wk, wq, att);

    // 3) softmax over m, in place
    softmax_kernel<<<dim3(NR_ * B_), 256, 0, stream>>>(att);

    // 4) att^T · v + residual -> d_out
    out_kernel<<<dim3((K_ / 16) * (DK_ / 16), B_, NR_), 32, 0, stream>>>(
        att, wv, f_a, out);
}